// MultiheadSimplifiedPrecisionAttentionBlock_87239375716640
// MI455X (gfx1250) — compile-verified
//
#include <hip/hip_runtime.h>
#include <hip/hip_bf16.h>
#include <math.h>

// Problem constants (match reference)
#define B_  2
#define S_  256
#define H_  8
#define DK_ 64
#define DV_ 64
#define DE_ 512
#define DT_ 512
#define EPS_ 1e-5f

typedef float v2f __attribute__((ext_vector_type(2)));
typedef float v8f __attribute__((ext_vector_type(8)));
typedef unsigned int u32x4 __attribute__((ext_vector_type(4)));
typedef int i32x8 __attribute__((ext_vector_type(8)));
typedef int i32x4 __attribute__((ext_vector_type(4)));

// ---------------------------------------------------------------------------
// CDNA5 fp32 tensor op: V_WMMA_F32_16X16X4_F32  (D = A(16x4) * B(4x16) + C)
// A frag: lane holds rows m=lane&15, a.x=K(kb), a.y=K(kb+1), kb=(lane>=16)?2:0
// B frag: lane holds col n=lane&15, b.x=B[kb][n], b.y=B[kb+1][n]
// C/D:    element v -> (m = v + 8*(lane>=16), n = lane&15)
// ---------------------------------------------------------------------------
__device__ __forceinline__ v8f wmma4(v2f a, v2f b, v8f c) {
  return __builtin_amdgcn_wmma_f32_16x16x4_f32(false, a, false, b, (short)0, c,
                                               false, false);
}

// ---------------------------------------------------------------------------
// TDM: issue a 2-D tensor_load_to_lds of `rows` x `width` fp32 elements with
// row stride `stride_elems`, landing compact at LDS byte offset `lds_off`.
// Descriptor bit layout per CDNA5 ISA ch.8 (D# groups 0/1; groups 2/3 zero
// for a 2-D tile; 5th operand group unused -> zeros). Tracked by TENSORcnt.
// This toolchain exposes the 6-arg builtin form.
// ---------------------------------------------------------------------------
__device__ __forceinline__ void tdm_load_2d(unsigned int lds_off,
                                            const float* gaddr, int width,
                                            int rows, int stride_elems) {
  unsigned long long ga = (unsigned long long)(uintptr_t)gaddr;
  u32x4 g0;
  g0.x = 1u;                                            // count=1 (valid D#)
  g0.y = lds_off;                                       // lds_addr
  g0.z = (unsigned int)ga;                              // global_addr[31:0]
  g0.w = (unsigned int)((ga >> 32) & 0x01FFFFFFu) |     // global_addr[56:32]
         (2u << 30);                                    // type=2 ("image")
  i32x8 g1;
  g1[0] = 0x00020000;                                   // data_size=4B, mask=0
  g1[1] = (width & 0xFFFF) << 16;                       // tensor_dim0[15:0]
  g1[2] = ((width >> 16) & 0xFFFF) | ((rows & 0xFFFF) << 16);   // td0 hi, td1 lo
  g1[3] = ((rows >> 16) & 0xFFFF) | ((width & 0xFFFF) << 16);   // td1 hi, tile_dim0
  g1[4] = rows & 0xFFFF;                                // tile_dim1 (tile_dim2=0)
  g1[5] = stride_elems;                                 // tensor_dim0_stride lo
  g1[6] = 0;                                            // stride hi, dim1_stride lo
  g1[7] = 0;
  i32x4 gz = {0, 0, 0, 0};
  i32x8 gz8 = {0, 0, 0, 0, 0, 0, 0, 0};
  __builtin_amdgcn_tensor_load_to_lds(g0, g1, gz, gz, gz8, 0);
}

// ---------------- workspace layout (float offsets) -------------------------
#define WOFF_EFR 0                 // S*DV*H = 131072, layout (s,d,h)
#define WOFF_EFI 131072
#define WOFF_EBR 262144
#define WOFF_EBI 393216
#define WOFF_VK2 524288            // S*H = 2048
#define WOFF_UQR 526336            // each U buffer: B*S*DT = 262144
#define WOFF_UQI 788480
#define WOFF_UKR 1050624
#define WOFF_UKI 1312768
#define WOFF_UVR 1574912
#define WOFF_UVI 1837056
#define WOFF_VR  2099200
#define WOFF_VI  2361344
#define WOFF_PR  2623488
#define WOFF_PI  2885632
#define WOFF_Q2  3147776           // B*S*H = 4096
#define WOFF_K2  3151872

// ---------------- output layout (float offsets, return order) --------------
#define OOFF_EST 0                 // est_latent (B,2,S,DV,H)      524288
#define OOFF_OUT 524288            // out        (B,2,S,DE)        524288
#define OOFF_QIJ 1048576           // Q_ij       (B,2,S,S,H)      2097152
#define OOFF_ZIJ 3145728           // Z_ij       (B,2,S,S,DT)   134217728
#define OOFF_LAM 137363456         // epoch_lambdas (2,H*DV,1)       1024

// ===========================================================================
// 1) phase factors ef/eb, Vk2 = |ef|^2, epoch_lambdas
// ===========================================================================
__global__ void k_phase(const float* __restrict__ tm,
                        const float* __restrict__ li,   // (H, DV/2)
                        const float* __restrict__ lr,   // (H,)
                        float* __restrict__ efr, float* __restrict__ efi,
                        float* __restrict__ ebr, float* __restrict__ ebi,
                        float* __restrict__ vk2,
                        float* __restrict__ lamOut) {
  int gid = blockIdx.x * 256 + threadIdx.x;       // gid = (s*64+d)*8+h
  int s = gid / (DV_ * H_);
  int r = gid % (DV_ * H_);
  int d = r / H_;
  int h = r % H_;
  float invt = 1.0f / (tm[S_ - 1] - tm[0]);
  float t = tm[s] * invt;
  float lim = (d < 32) ? li[h * 32 + d] : -li[h * 32 + (d - 32)];
  float ph = t * lim;
  float mg = expf(t * lr[h]);
  float c = cosf(ph), sn = sinf(ph);
  efr[gid] = mg * c;
  efi[gid] = mg * sn;
  ebr[gid] = c / mg;
  ebi[gid] = -sn / mg;
  if (d == 0) vk2[s * H_ + h] = mg * mg;
  if (gid < 2 * H_ * DV_) {  // epoch_lambdas[c, h*DV + d]
    int c2 = gid >> 9, rest = gid & 511;
    int h2 = rest >> 6, d2 = rest & 63;
    float val = (c2 == 0) ? lr[h2]
                          : ((d2 < 32) ? li[h2 * 32 + d2] : -li[h2 * 32 + d2 - 32]);
    lamOut[gid] = val;
  }
}

// ===========================================================================
// 2) Complex GEMM via fp32 WMMA with TDM-staged A operand:
//    Y = (Xr+iXi)(W0+iW1) + (b0+ib1)
//    The 8 waves of a block share one 16-row A tile; TDM DMAs the (16 x 256)
//    fp32 chunk of Xr and Xi into LDS once per block per K-chunk, fenced by
//    s_wait_tensorcnt + workgroup barrier. Optional epilogues: U = eb*Y
//    (projections), raw copy (V), plain (output projection).
//    Requires: (N/16) multiple of 8, 16-row tiles don't cross batch rows.
// ===========================================================================
#define KC_ 256   // K chunk staged in LDS: 2 * 16 * KC_ * 4B = 32 KB
__global__ __launch_bounds__(256)
void k_cgemm(const float* __restrict__ Xr, const float* __restrict__ Xi,
             long xBS, int ldx,
             const float* __restrict__ W, int Kdim, int N,
             const float* __restrict__ bias,
             float* __restrict__ Yr, float* __restrict__ Yi,
             long yBS, int ldy,
             const float* __restrict__ ebr, const float* __restrict__ ebi,
             float* __restrict__ rawR, float* __restrict__ rawI) {
  extern __shared__ float lA[];               // [2][16][KC_], at LDS offset 0
  const int lane = threadIdx.x & 31;
  const int wave = threadIdx.x >> 5;
  const int nTiles = N >> 4;
  const int tileId = blockIdx.x * 8 + wave;
  const int m0 = ((blockIdx.x * 8) / nTiles) << 4;   // shared by all 8 waves
  const int n0 = (tileId % nTiles) << 4;
  const int ml = lane & 15;
  const int kb = (lane >> 4) << 1;

  const int bA = m0 / S_, sA = m0 % S_;
  const float* xrTile = Xr + (long)bA * xBS + (long)sA * ldx;
  const float* xiTile = Xi + (long)bA * xBS + (long)sA * ldx;
  const int n = n0 + ml;
  const float* W0 = W;
  const float* W1 = W + (long)Kdim * N;

  const float* lr_ = lA + ml * KC_;           // Xr row (this lane's A row)
  const float* li_ = lA + 16 * KC_ + ml * KC_;// Xi row

  v8f accR = {}, accI = {};
  for (int kc = 0; kc < Kdim; kc += KC_) {
    if (wave == 0) {                          // one TDM issue per block
      tdm_load_2d(0u, xrTile + kc, KC_, 16, ldx);
      tdm_load_2d((unsigned int)(16 * KC_ * 4), xiTile + kc, KC_, 16, ldx);
      __builtin_amdgcn_s_wait_tensorcnt(0);
    }
    __syncthreads();                          // LDS chunk visible to all waves
    for (int k = 0; k < KC_; k += 4) {
      int kk = k + kb;
      int kg = kc + kk;
      v2f ar, ai, b0, b1;
      ar.x = lr_[kk];               ar.y = lr_[kk + 1];
      ai.x = li_[kk];               ai.y = li_[kk + 1];
      b0.x = W0[(long)kg * N + n];  b0.y = W0[(long)(kg + 1) * N + n];
      b1.x = W1[(long)kg * N + n];  b1.y = W1[(long)(kg + 1) * N + n];
      accR = wmma4(ar, b0, accR);
      accR = wmma4(-ai, b1, accR);  // f32 WMMA has no A-neg modifier: VALU negate
      accI = wmma4(ar, b1, accI);
      accI = wmma4(ai, b0, accI);
    }
    __syncthreads();                          // chunk consumed before overwrite
  }

  const float bR = bias[n];
  const float bI = bias[N + n];
  const int mhi = (lane >> 4) << 3;
  const int d = n & 63, h = n >> 6;   // only meaningful when N==DT (projections)
#pragma unroll
  for (int v = 0; v < 8; ++v) {
    int row = m0 + v + mhi;
    int b = row / S_, s = row % S_;
    float yr = accR[v] + bR;
    float yi = accI[v] + bI;
    long oy = (long)b * yBS + (long)s * ldy + n;
    if (rawR) { rawR[oy] = yr; rawI[oy] = yi; }
    if (ebr) {
      float er = ebr[(s * 64 + d) * 8 + h];
      float ei = ebi[(s * 64 + d) * 8 + h];
      Yr[oy] = er * yr - ei * yi;
      Yi[oy] = er * yi + ei * yr;
    } else {
      Yr[oy] = yr;
      Yi[oy] = yi;
    }
  }
}

// ===========================================================================
// 3) per-(b,s,h) sum over d of |Uq|^2 and |Uk|^2
// ===========================================================================
__global__ void k_sumsq(const float* __restrict__ Uqr, const float* __restrict__ Uqi,
                        const float* __restrict__ Ukr, const float* __restrict__ Uki,
                        float* __restrict__ Q2, float* __restrict__ K2) {
  int gid = blockIdx.x * 256 + threadIdx.x;  // B*S*H
  int bs = gid / H_, h = gid % H_;
  const float* qr = Uqr + (long)bs * DT_ + h * 64;
  const float* qi = Uqi + (long)bs * DT_ + h * 64;
  const float* kr = Ukr + (long)bs * DT_ + h * 64;
  const float* ki = Uki + (long)bs * DT_ + h * 64;
  float sq = 0.f, sk = 0.f;
  for (int d = 0; d < 64; ++d) {
    float a = qr[d], b = qi[d]; sq += a * a + b * b;
    float c = kr[d], e = ki[d]; sk += c * c + e * e;
  }
  Q2[gid] = sq;
  K2[gid] = sk;
}

// ===========================================================================
// 4) scores: WMMA cross = Uq_r·Uk_r^T + Uq_i·Uk_i^T, fused V_avg/log epilogue.
//    Written in place into the Q_ij real region (b,0,i,j,h).
// ===========================================================================
__global__ __launch_bounds__(256)
void k_scores(const float* __restrict__ Uqr, const float* __restrict__ Uqi,
              const float* __restrict__ Ukr, const float* __restrict__ Uki,
              const float* __restrict__ vk2,
              const float* __restrict__ Q2, const float* __restrict__ K2,
              const float* __restrict__ tm,
              const float* __restrict__ lr, const float* __restrict__ lo,
              const float* __restrict__ lg, const float* __restrict__ nf,
              const float* __restrict__ tau,
              float* __restrict__ qout) {
  const int lane = threadIdx.x & 31;
  const int wave = threadIdx.x >> 5;
  const int tileId = blockIdx.x * 8 + wave;     // 4096 tiles
  const int bh = tileId >> 8, t2 = tileId & 255;
  const int b = bh >> 3, h = bh & 7;
  const int i0 = (t2 >> 4) << 4, j0 = (t2 & 15) << 4;
  const int ml = lane & 15;
  const int kb = (lane >> 4) << 1;

  const float* qr = Uqr + (long)(b * S_ + i0 + ml) * DT_ + h * 64;
  const float* qi = Uqi + (long)(b * S_ + i0 + ml) * DT_ + h * 64;
  const float* kr = Ukr + (long)(b * S_ + j0 + ml) * DT_ + h * 64;
  const float* ki = Uki + (long)(b * S_ + j0 + ml) * DT_ + h * 64;

  v8f acc = {};
  for (int k = 0; k < DK_; k += 4) {
    int kk = k + kb;
    v2f aqr, aqi, bkr, bki;
    aqr.x = qr[kk]; aqr.y = qr[kk + 1];
    aqi.x = qi[kk]; aqi.y = qi[kk + 1];
    bkr.x = kr[kk]; bkr.y = kr[kk + 1];
    bki.x = ki[kk]; bki.y = ki[kk + 1];
    acc = wmma4(aqr, bkr, acc);
    acc = wmma4(aqi, bki, acc);
  }

  const float invt = 1.0f / (tm[S_ - 1] - tm[0]);
  const float lamre = lr[h];
  const float den = 2.0f * lamre;
  const float o2 = lo[h] * lo[h];
  const float g2 = lg[h] * lg[h] + EPS_;
  const float nf2 = nf[h] * nf[h] + EPS_;
  const float tau2 = tau[h] * tau[h];
  const bool denOk = fabsf(den) > 1e-6f;
  const int j = j0 + ml;
  const float tj = tm[j] * invt;
  const float k2 = K2[(b * S_ + j) * H_ + h];
  const int mhi = (lane >> 4) << 3;
#pragma unroll
  for (int v = 0; v < 8; ++v) {
    int i = i0 + v + mhi;
    float ti = tm[i] * invt;
    float dtv = fmaxf(ti - tj, 0.0f);
    float x = dtv * den;
    float g = denOk ? (expm1f(x) / den) : dtv;
    float Vav = o2 * g + g2 * expf(x);
    float R = vk2[i * H_ + h] *
              (k2 + Q2[(b * S_ + i) * H_ + h] - 2.0f * acc[v]);
    float base = nf2 * Vav + R;
    float sc = -tau2 * logf(base);
    qout[((((long)(b * 2) * S_ + i) * S_) + j) * H_ + h] = sc;
  }
}

// ===========================================================================
// 5) causal softmax over j, in place in Q_ij real; zeros the imag half.
// ===========================================================================
__global__ void k_softmax(float* __restrict__ q) {
  __shared__ float red[256];
  int rid = blockIdx.x;                       // ((b*H + h)*S + i)
  int b = rid / (H_ * S_);
  int r2 = rid % (H_ * S_);
  int h = r2 / S_, i = r2 % S_;
  int j = threadIdx.x;
  long idxR = ((((long)(b * 2 + 0) * S_ + i) * S_) + j) * H_ + h;
  long idxI = ((((long)(b * 2 + 1) * S_ + i) * S_) + j) * H_ + h;
  float v = (j <= i) ? q[idxR] : -__builtin_inff();
  red[j] = v; __syncthreads();
  for (int s = 128; s > 0; s >>= 1) {
    if (j < s) red[j] = fmaxf(red[j], red[j + s]);
    __syncthreads();
  }
  float mx = red[0]; __syncthreads();
  float e = (j <= i) ? expf(v - mx) : 0.0f;
  red[j] = e; __syncthreads();
  for (int s = 128; s > 0; s >>= 1) {
    if (j < s) red[j] += red[j + s];
    __syncthreads();
  }
  float inv = 1.0f / red[0];
  q[idxR] = e * inv;
  q[idxI] = 0.0f;
}

// ===========================================================================
// 6) er/ei = attn @ Uv (WMMA), fused est/eta/el/me epilogue:
//    writes est_latent output and P (= me * el) for the output projection.
// ===========================================================================
__global__ __launch_bounds__(256)
void k_av(const float* __restrict__ attn,   // Q_ij real region
          const float* __restrict__ Uvr, const float* __restrict__ Uvi,
          const float* __restrict__ efr, const float* __restrict__ efi,
          const float* __restrict__ Vr, const float* __restrict__ Vi,
          const float* __restrict__ etap,
          float* __restrict__ estOut,
          float* __restrict__ Pr, float* __restrict__ Pi) {
  const int lane = threadIdx.x & 31;
  const int wave = threadIdx.x >> 5;
  const int tileId = blockIdx.x * 8 + wave;   // 1024 tiles: 16 bh * (16 x 4)
  const int bh = tileId >> 6, t2 = tileId & 63;
  const int b = bh >> 3, h = bh & 7;
  const int i0 = (t2 >> 2) << 4, d0 = (t2 & 3) << 4;
  const int ml = lane & 15;
  const int kb = (lane >> 4) << 1;

  const float* aRow = attn + (((long)(b * 2) * S_ + (i0 + ml)) * S_) * H_ + h;
  const float* br = Uvr + (long)b * S_ * DT_ + h * 64 + (d0 + ml);
  const float* bi = Uvi + (long)b * S_ * DT_ + h * 64 + (d0 + ml);

  v8f accR = {}, accI = {};
  for (int k = 0; k < S_; k += 4) {
    int kk = k + kb;
    v2f a, b2r, b2i;
    a.x   = aRow[(long)kk * H_];        a.y   = aRow[(long)(kk + 1) * H_];
    b2r.x = br[(long)kk * DT_];         b2r.y = br[(long)(kk + 1) * DT_];
    b2i.x = bi[(long)kk * DT_];         b2i.y = bi[(long)(kk + 1) * DT_];
    accR = wmma4(a, b2r, accR);
    accI = wmma4(a, b2i, accI);
  }

  const int d = d0 + ml;
  const int n = h * 64 + d;
  float ep  = etap[d * H_ + h];
  float eta = 1.0f / (1.0f + expf(-ep));
  float se  = 1.0f / (1.0f + expf(-eta));
  float omn = 1.0f - eta;
  float mer = efr[d * 8 + h];           // ef at s=0
  float mei = efi[d * 8 + h];
  const int mhi = (lane >> 4) << 3;
#pragma unroll
  for (int v = 0; v < 8; ++v) {
    int i = i0 + v + mhi;
    float er_ = accR[v], ei_ = accI[v];
    float fr = efr[(i * 64 + d) * 8 + h];
    float fi = efi[(i * 64 + d) * 8 + h];
    float estr = fr * er_ - fi * ei_;
    float esti = fr * ei_ + fi * er_;
    long ov = (long)(b * S_ + i) * DT_ + n;
    float vr = Vr[ov], vi = Vi[ov];
    float elr = omn * vr + se * estr;
    float eli = omn * vi + se * esti;
    estOut[(((long)(b * 2 + 0) * S_ + i) * DV_ + d) * H_ + h] = elr;
    estOut[(((long)(b * 2 + 1) * S_ + i) * DV_ + d) * H_ + h] = eli;
    Pr[ov] = mer * elr - mei * eli;
    Pi[ov] = mer * eli + mei * elr;
  }
}

// ===========================================================================
// 7) Z_ij = ef(i) * Uv(j): 512 MB streaming output -> non-temporal stores.
// ===========================================================================
__global__ void k_zij(const float* __restrict__ Uvr, const float* __restrict__ Uvi,
                      const float* __restrict__ efr, const float* __restrict__ efi,
                      float* __restrict__ z) {
  int n = threadIdx.x;                       // 0..511
  long blk = blockIdx.x;                     // b*S*S + i*S + j
  int b = (int)(blk / (S_ * S_));
  int r = (int)(blk % (S_ * S_));
  int i = r / S_, j = r % S_;
  int d = n & 63, h = n >> 6;
  float fr = efr[(i * 64 + d) * 8 + h];
  float fi = efi[(i * 64 + d) * 8 + h];
  long ov = (long)(b * S_ + j) * DT_ + n;
  float ur = Uvr[ov], ui = Uvi[ov];
  float zr = fr * ur - fi * ui;
  float zi = fr * ui + fi * ur;
  long o0 = ((((long)(b * 2 + 0) * S_ + i) * S_) + j) * DT_ + n;
  long o1 = ((((long)(b * 2 + 1) * S_ + i) * S_) + j) * DT_ + n;
  __builtin_nontemporal_store(zr, z + o0);
  __builtin_nontemporal_store(zi, z + o1);
}

// ===========================================================================
extern "C" void kernel_launch(void* const* d_in, const int* in_sizes, int n_in,
                              void* d_out, int out_size, void* d_ws, size_t ws_size,
                              hipStream_t stream) {
  const float* Zq  = (const float*)d_in[0];
  const float* Zk  = (const float*)d_in[1];
  const float* Zv  = (const float*)d_in[2];
  const float* tm  = (const float*)d_in[3];
  const float* Wq  = (const float*)d_in[4];
  const float* bq  = (const float*)d_in[5];
  const float* Wk  = (const float*)d_in[6];
  const float* bk  = (const float*)d_in[7];
  const float* Wv  = (const float*)d_in[8];
  const float* bv  = (const float*)d_in[9];
  const float* Wp  = (const float*)d_in[10];
  const float* bp  = (const float*)d_in[11];
  const float* li  = (const float*)d_in[12];
  const float* lr  = (const float*)d_in[13];
  const float* lo  = (const float*)d_in[14];
  const float* lg  = (const float*)d_in[15];
  const float* nf  = (const float*)d_in[16];
  const float* tau = (const float*)d_in[17];
  const float* etp = (const float*)d_in[18];

  float* ws  = (float*)d_ws;
  float* out = (float*)d_out;

  float* efr = ws + WOFF_EFR;  float* efi = ws + WOFF_EFI;
  float* ebr = ws + WOFF_EBR;  float* ebi = ws + WOFF_EBI;
  float* vk2 = ws + WOFF_VK2;
  float* Uqr = ws + WOFF_UQR;  float* Uqi = ws + WOFF_UQI;
  float* Ukr = ws + WOFF_UKR;  float* Uki = ws + WOFF_UKI;
  float* Uvr = ws + WOFF_UVR;  float* Uvi = ws + WOFF_UVI;
  float* Vr  = ws + WOFF_VR;   float* Vi  = ws + WOFF_VI;
  float* Pr  = ws + WOFF_PR;   float* Pi  = ws + WOFF_PI;
  float* Q2  = ws + WOFF_Q2;   float* K2  = ws + WOFF_K2;

  const size_t ldsA = 2u * 16u * KC_ * sizeof(float);   // 32 KB TDM staging

  // 1) phase factors + epoch_lambdas
  k_phase<<<(S_ * DV_ * H_) / 256, 256, 0, stream>>>(tm, li, lr, efr, efi, ebr,
                                                     ebi, vk2, out + OOFF_LAM);

  // 2) projections (complex WMMA GEMM, TDM-staged A, fused bias + eb rotation)
  const long xBS = (long)2 * S_ * DE_;     // Z layout (B,2,S,DE)
  const long uBS = (long)S_ * DT_;         // U layout (B*S, DT)
  k_cgemm<<<128, 256, ldsA, stream>>>(Zq, Zq + (long)S_ * DE_, xBS, DE_, Wq,
                                      DE_, DT_, bq, Uqr, Uqi, uBS, DT_, ebr,
                                      ebi, nullptr, nullptr);
  k_cgemm<<<128, 256, ldsA, stream>>>(Zk, Zk + (long)S_ * DE_, xBS, DE_, Wk,
                                      DE_, DT_, bk, Ukr, Uki, uBS, DT_, ebr,
                                      ebi, nullptr, nullptr);
  k_cgemm<<<128, 256, ldsA, stream>>>(Zv, Zv + (long)S_ * DE_, xBS, DE_, Wv,
                                      DE_, DT_, bv, Uvr, Uvi, uBS, DT_, ebr,
                                      ebi, Vr, Vi);

  // 3) sum over d of |Uq|^2, |Uk|^2
  k_sumsq<<<(B_ * S_ * H_) / 256, 256, 0, stream>>>(Uqr, Uqi, Ukr, Uki, Q2, K2);

  // 4) scores (WMMA cross + fused V_avg/log), staged in Q_ij real region
  k_scores<<<512, 256, 0, stream>>>(Uqr, Uqi, Ukr, Uki, vk2, Q2, K2, tm, lr,
                                    lo, lg, nf, tau, out + OOFF_QIJ);

  // 5) causal softmax (in place) + zero Q_ij imag part
  k_softmax<<<B_ * H_ * S_, 256, 0, stream>>>(out + OOFF_QIJ);

  // 6) attn @ Uv (WMMA) + est/el/P epilogue (writes est_latent)
  k_av<<<128, 256, 0, stream>>>(out + OOFF_QIJ, Uvr, Uvi, efr, efi, Vr, Vi,
                                etp, out + OOFF_EST, Pr, Pi);

  // 7) output projection straight into `out` (B,2,S,DE)
  k_cgemm<<<128, 256, ldsA, stream>>>(Pr, Pi, (long)S_ * DT_, DT_, Wp, DT_,
                                      DE_, bp, out + OOFF_OUT,
                                      out + OOFF_OUT + (long)S_ * DE_,
                                      (long)2 * S_ * DE_, DE_, nullptr,
                                      nullptr, nullptr, nullptr);

  // 8) Z_ij: 512 MB streaming write, non-temporal
  k_zij<<<B_ * S_ * S_, 512, 0, stream>>>(Uvr, Uvi, efr, efi, out + OOFF_ZIJ);
}